// ElasticHead_76716705841923
// MI455X (gfx1250) — compile-verified
//
#include <hip/hip_runtime.h>
#include <math.h>

typedef float v2f __attribute__((ext_vector_type(2)));
typedef float v8f __attribute__((ext_vector_type(8)));

#define HWs   6400
#define Wd    80
#define Hd    80
#define C1c   256
#define C2c   256
#define KKtot 2304   // C1*9
#define KCH2  384    // K chunk staged in LDS (6 chunks)
#define BP2   386    // LDS column stride in floats (bank spread: 386 % 64 == 2)

#define AS1P __attribute__((address_space(1))) int*
#define AS3P __attribute__((address_space(3))) int*

#if __has_builtin(__builtin_amdgcn_global_load_async_to_lds_b32)
#define HAS_ASYNC_LDS 1
#else
#define HAS_ASYNC_LDS 0
#endif

__device__ __forceinline__ void wait_async_zero() {
#if HAS_ASYNC_LDS
#if __has_builtin(__builtin_amdgcn_s_wait_asynccnt)
  __builtin_amdgcn_s_wait_asynccnt(0);
#else
  asm volatile("s_wait_asynccnt 0x0" ::: "memory");
#endif
#endif
}

// ---------------------------------------------------------------------------
// Kernel 1: offset conv  (B,27,H,W) = conv3x3(x, w_off) + b_off
// Block = 256 threads = 16x16 spatial tile; weights staged in 64-channel chunks.
// ---------------------------------------------------------------------------
__global__ __launch_bounds__(256) void k_offconv(const float* __restrict__ x,
                                                 const float* __restrict__ w_off,
                                                 const float* __restrict__ b_off,
                                                 float* __restrict__ off_out) {
  __shared__ float wsl[64 * 243];          // 62,208 B
  const int t = threadIdx.x;
  const int tile = blockIdx.x;             // 0..24
  const int b = blockIdx.y;                // 0..3
  const int ty = tile / 5, tx = tile % 5;
  const int lh = t >> 4, lw = t & 15;
  const int h = ty * 16 + lh, w = tx * 16 + lw;

  float acc[27];
#pragma unroll
  for (int o = 0; o < 27; ++o) acc[o] = 0.f;

  const float* xb = x + (size_t)b * C1c * HWs;

  for (int cb = 0; cb < 4; ++cb) {
    __syncthreads();
    for (int i = t; i < 64 * 243; i += 256) {
      int cc = i / 243, r = i - cc * 243;   // r = oc*9 + p
      int oc = r / 9, p = r - oc * 9;
      wsl[i] = w_off[((size_t)oc * C1c + (cb * 64 + cc)) * 9 + p];
    }
    __syncthreads();
    for (int c64 = 0; c64 < 64; ++c64) {
      const int c = cb * 64 + c64;
      float xv[9];
#pragma unroll
      for (int ki = 0; ki < 3; ++ki)
#pragma unroll
        for (int kj = 0; kj < 3; ++kj) {
          int yy = h + ki - 1, xx = w + kj - 1;
          bool ok = (yy >= 0) && (yy < Hd) && (xx >= 0) && (xx < Wd);
          xv[ki * 3 + kj] = ok ? xb[(size_t)c * HWs + yy * Wd + xx] : 0.f;
        }
      const float* wp = &wsl[c64 * 243];
#pragma unroll
      for (int o = 0; o < 27; ++o) {
        float s = acc[o];
#pragma unroll
        for (int p = 0; p < 9; ++p) s = fmaf(xv[p], wp[o * 9 + p], s);
        acc[o] = s;
      }
    }
  }
#pragma unroll
  for (int o = 0; o < 27; ++o)
    off_out[((size_t)(b * 27 + o)) * HWs + h * Wd + w] = acc[o] + b_off[o];
}

// ---------------------------------------------------------------------------
// Kernel 2: deformable bilinear sampling -> patches[k = c*9+p][hw] (one batch)
// Offsets interleaved (dy,dx): dy[p]=ch 2p, dx[p]=ch 2p+1, mask=ch 18+p.
// Mask folded into the 4 bilinear weights; one pass over all 256 channels.
// ---------------------------------------------------------------------------
__global__ __launch_bounds__(256) void k_sample(const float* __restrict__ x,
                                                const float* __restrict__ off_out,
                                                float* __restrict__ patches, int b) {
  const int gid = blockIdx.x * 256 + threadIdx.x;   // < 9*6400
  const int p = gid / HWs, hw = gid - p * HWs;
  const int h = hw / Wd, w = hw - h * Wd;

  const float* ob = off_out + (size_t)b * 27 * HWs;
  const float dy = ob[(size_t)(2 * p) * HWs + hw];
  const float dx = ob[(size_t)(2 * p + 1) * HWs + hw];
  const float mk = ob[(size_t)(18 + p) * HWs + hw];
  const float mask = 1.f / (1.f + expf(-mk));

  const float ys = dy + (float)(h - 1 + p / 3);
  const float xs = dx + (float)(w - 1 + p % 3);
  const float y0f = floorf(ys), x0f = floorf(xs);
  const float wy = ys - y0f, wx = xs - x0f;
  const int y0 = (int)y0f, x0 = (int)x0f, y1 = y0 + 1, x1 = x0 + 1;

  const float v00 = (y0 >= 0 && y0 < Hd && x0 >= 0 && x0 < Wd) ? 1.f : 0.f;
  const float v01 = (y0 >= 0 && y0 < Hd && x1 >= 0 && x1 < Wd) ? 1.f : 0.f;
  const float v10 = (y1 >= 0 && y1 < Hd && x0 >= 0 && x0 < Wd) ? 1.f : 0.f;
  const float v11 = (y1 >= 0 && y1 < Hd && x1 >= 0 && x1 < Wd) ? 1.f : 0.f;

  const int cy0 = min(max(y0, 0), Hd - 1), cy1 = min(max(y1, 0), Hd - 1);
  const int cx0 = min(max(x0, 0), Wd - 1), cx1 = min(max(x1, 0), Wd - 1);
  const int i00 = cy0 * Wd + cx0, i01 = cy0 * Wd + cx1;
  const int i10 = cy1 * Wd + cx0, i11 = cy1 * Wd + cx1;

  const float w00 = (1.f - wy) * (1.f - wx) * mask * v00;
  const float w01 = (1.f - wy) * wx * mask * v01;
  const float w10 = wy * (1.f - wx) * mask * v10;
  const float w11 = wy * wx * mask * v11;

  const float* xb = x + (size_t)b * C1c * HWs;
#pragma unroll 4
  for (int c = 0; c < C1c; ++c) {
    const float* xc = xb + (size_t)c * HWs;
    float v = w00 * xc[i00] + w01 * xc[i01] + w10 * xc[i10] + w11 * xc[i11];
    patches[((size_t)(c * 9 + p)) * HWs + hw] = v;
  }
}

// ---------------------------------------------------------------------------
// Kernel 3: fp32 WMMA GEMM  out[256,6400] = W[256,2304] @ patches[2304,6400]
// Block = 512 threads (16 waves) on a 32-pixel column stripe.
// Each wave: one 16-row M-tile x two 16-col N-tiles -> each A b64 load feeds
// two v_wmma_f32_16x16x4_f32 (halves weight traffic per FLOP vs 1 N-tile).
// B panel (32 cols x 384 K) staged in LDS column-major, stride 386 floats.
// Staging uses GLOBAL_LOAD_ASYNC_TO_LDS (builtin present on this toolchain).
// ---------------------------------------------------------------------------
__global__ __launch_bounds__(512) void k_gemm(const float* __restrict__ patches,
                                              const float* __restrict__ weight,
                                              const float* __restrict__ bias,
                                              float* __restrict__ conv_out, int b) {
  __shared__ float bpan[32 * BP2];         // 49,408 B
  const int t = threadIdx.x;
  const int lane = t & 31, wave = t >> 5;  // 16 waves
  const int nl = lane & 15, half = lane >> 4;
  const int col0 = blockIdx.x * 32;
  const int m0 = wave * 16;                // one 16-row M-tile per wave

  v8f c0 = {0.f, 0.f, 0.f, 0.f, 0.f, 0.f, 0.f, 0.f};
  v8f c1 = {0.f, 0.f, 0.f, 0.f, 0.f, 0.f, 0.f, 0.f};

  for (int kb = 0; kb < KKtot; kb += KCH2) {
    __syncthreads();   // previous chunk fully consumed
    {
      const int cc = t & 31, ks = t >> 5;   // 32 cols x 16 k-phases
      for (int k = ks; k < KCH2; k += 16) {
        const size_t gidx = (size_t)(kb + k) * HWs + col0 + cc;
        const int lidx = cc * BP2 + k;
#if HAS_ASYNC_LDS
        float* gp = const_cast<float*>(patches) + gidx;
        __builtin_amdgcn_global_load_async_to_lds_b32(
            (AS1P)gp, (AS3P)&bpan[lidx], 0, 0);
#else
        bpan[lidx] = patches[gidx];
#endif
      }
    }
    wait_async_zero();
    __syncthreads();   // panel visible to all waves

    const float* wr = weight + (size_t)(m0 + nl) * KKtot + kb + 2 * half;
    const int bb0 = nl * BP2 + 2 * half;
    const int bb1 = (nl + 16) * BP2 + 2 * half;

#pragma unroll 4
    for (int kk = 0; kk < KCH2; kk += 4) {
      // A 16x4 f32 layout: lanes 0-15 K={kk,kk+1}; lanes 16-31 K={kk+2,kk+3}
      v2f av = *(const v2f*)(wr + kk);
      v2f bv0 = *(const v2f*)&bpan[bb0 + kk];
      v2f bv1 = *(const v2f*)&bpan[bb1 + kk];
      c0 = __builtin_amdgcn_wmma_f32_16x16x4_f32(false, av, false, bv0,
                                                 (short)0, c0, false, false);
      c1 = __builtin_amdgcn_wmma_f32_16x16x4_f32(false, av, false, bv1,
                                                 (short)0, c1, false, false);
    }
  }

  float* outb = conv_out + (size_t)b * C2c * HWs;
#pragma unroll
  for (int r = 0; r < 8; ++r) {
    // D layout: VGPR r -> M = r + 8*half, N = nl
    const int row = m0 + r + 8 * half;
    const float bi = bias[row];
    outb[(size_t)row * HWs + col0 + nl] = c0[r] + bi;
    outb[(size_t)row * HWs + col0 + 16 + nl] = c1[r] + bi;
  }
}

// ---------------------------------------------------------------------------
// Kernel 4: per-channel BN stats over (B,H,W) -> fused scale/shift
// ---------------------------------------------------------------------------
__global__ __launch_bounds__(256) void k_stats(const float* __restrict__ conv_out,
                                               const float* __restrict__ gamma,
                                               const float* __restrict__ beta,
                                               float* __restrict__ chS,
                                               float* __restrict__ chT) {
  __shared__ float s1[256], s2[256];
  const int ch = blockIdx.x, t = threadIdx.x;
  float sum = 0.f, sq = 0.f;
  for (int b = 0; b < 4; ++b) {
    const float* p = conv_out + (size_t)(b * C2c + ch) * HWs;
    for (int i = t; i < HWs; i += 256) {
      float v = p[i];
      sum += v;
      sq = fmaf(v, v, sq);
    }
  }
  s1[t] = sum; s2[t] = sq;
  __syncthreads();
  for (int s = 128; s > 0; s >>= 1) {
    if (t < s) { s1[t] += s1[t + s]; s2[t] += s2[t + s]; }
    __syncthreads();
  }
  if (t == 0) {
    const float n = 4.f * (float)HWs;
    const float mean = s1[0] / n;
    const float var = s2[0] / n - mean * mean;
    const float sc = gamma[ch] * rsqrtf(var + 1e-5f);
    chS[ch] = sc;
    chT[ch] = beta[ch] - mean * sc;
  }
}

// ---------------------------------------------------------------------------
// Kernel 5: apply BN + SiLU, float4-vectorized
// ---------------------------------------------------------------------------
__global__ __launch_bounds__(256) void k_bnact(const float* __restrict__ conv_out,
                                               const float* __restrict__ chS,
                                               const float* __restrict__ chT,
                                               float* __restrict__ out) {
  const int i4 = blockIdx.x * 256 + threadIdx.x;  // < 1,638,400
  const int ch = (i4 / 1600) & 255;               // 1600 float4 per (b,ch)
  const float4 v = ((const float4*)conv_out)[i4];
  const float s = chS[ch], tt = chT[ch];
  float4 y;
  y.x = fmaf(v.x, s, tt);
  y.y = fmaf(v.y, s, tt);
  y.z = fmaf(v.z, s, tt);
  y.w = fmaf(v.w, s, tt);
  float4 r;
  r.x = y.x / (1.f + expf(-y.x));
  r.y = y.y / (1.f + expf(-y.y));
  r.z = y.z / (1.f + expf(-y.z));
  r.w = y.w / (1.f + expf(-y.w));
  ((float4*)out)[i4] = r;
}

// ---------------------------------------------------------------------------
extern "C" void kernel_launch(void* const* d_in, const int* in_sizes, int n_in,
                              void* d_out, int out_size, void* d_ws, size_t ws_size,
                              hipStream_t stream) {
  (void)in_sizes; (void)n_in; (void)out_size; (void)ws_size;
  const float* x      = (const float*)d_in[0];
  const float* w_off  = (const float*)d_in[1];
  const float* b_off  = (const float*)d_in[2];
  const float* weight = (const float*)d_in[3];
  const float* bias   = (const float*)d_in[4];
  const float* gamma  = (const float*)d_in[5];
  const float* beta   = (const float*)d_in[6];
  float* out = (float*)d_out;

  float* wsf      = (float*)d_ws;
  float* off_out  = wsf;               //   691,200 floats
  float* chS      = wsf + 691200;      //       256
  float* chT      = wsf + 691456;      //       256
  float* conv_out = wsf + 691712;      // 6,553,600
  float* patches  = wsf + 7245312;     // 14,745,600 (single-batch buffer)

  k_offconv<<<dim3(25, 4), 256, 0, stream>>>(x, w_off, b_off, off_out);
  for (int b = 0; b < 4; ++b) {
    k_sample<<<225, 256, 0, stream>>>(x, off_out, patches, b);
    k_gemm<<<200, 512, 0, stream>>>(patches, weight, bias, conv_out, b);
  }
  k_stats<<<256, 256, 0, stream>>>(conv_out, gamma, beta, chS, chT);
  k_bnact<<<6400, 256, 0, stream>>>(conv_out, chS, chT, out);
}